// CGCNN_Net_22548578304898
// MI455X (gfx1250) — compile-verified
//
#include <hip/hip_runtime.h>
#include <hip/hip_bf16.h>

// ---------------------------------------------------------------------------
// CGCNN on MI455X (gfx1250): fp32 WMMA (V_WMMA_F32_16X16X4_F32) GEMM core.
// B staged in LDS in fragment-ready interleaved layout so each WMMA B operand
// is a single aligned ds_load_b64 (no packing movs). A pitch 20 / B pair-row
// pitch 160 give conflict-free 64-bank LDS fragment reads.
// ---------------------------------------------------------------------------

typedef __attribute__((ext_vector_type(2))) float v2f;
typedef __attribute__((ext_vector_type(8))) float v8f;

#define BM 64
#define BN 64
#define BK 16
#define A_PITCH 20        // 16+4 pad: 20*m mod 64 distinct for m<16; pairs stay aligned
#define B_PPITCH 160      // dwords per K-pair row; 160 mod 64 == 32 -> lanes 16-31 use banks 32..63

// C = epilogue(alpha * A(MxK) @ B(KxN) + beta*P + bias), row-major, shapes
// must be multiples of BM/BN/BK (true for every call site here).
__global__ __launch_bounds__(128)
void gemm_wmma_f32(const float* __restrict__ A, int lda,
                   const float* __restrict__ B, int ldb,
                   const float* __restrict__ P,     // prev (may be null), stride N
                   const float* __restrict__ bias,  // may be null, length N
                   float* __restrict__ C,
                   int M, int N, int K,
                   float alpha, float beta, int relu)
{
    __shared__ __align__(16) float Alds[BM * A_PITCH];       // [row][k]
    __shared__ __align__(16) float Blds[8 * B_PPITCH];       // [k>>1][n*2 + (k&1)]

    const int tid  = threadIdx.x;
    const int lane = tid & 31;
    const int wave = tid >> 5;          // 0..3 -> 16-row strip of the 64x64 block tile
    const int bm   = blockIdx.y * BM;
    const int bn   = blockIdx.x * BN;

    const int mf = lane & 15;           // M index within frag (A), N index (B/C)
    const int kp = (lane >> 4) << 1;    // lanes 0-15 hold K={0,1}; lanes 16-31 K={2,3}

    // Per-thread staging coordinates (fixed across K loop)
    const int ar0 = tid >> 2;                 // A: rows tid>>2 and +32
    const int ac  = (tid & 3) << 2;
    const int bpr = tid >> 4;                 // B: pair-row 0..7
    const int bc  = (tid & 15) << 2;          // n base (4 cols)

    v8f acc[4] = {};                    // wave covers 16 rows x 64 cols (4 n-tiles)

    for (int k0 = 0; k0 < K; k0 += BK) {
        // ---- Stage A tile (64x16): 2 float4 per thread ----
        #pragma unroll
        for (int i = 0; i < 2; ++i) {
            int r = ar0 + i * 32;
            float4 av = *(const float4*)(A + (size_t)(bm + r) * lda + k0 + ac);
            *(float4*)(Alds + r * A_PITCH + ac) = av;
        }
        // ---- Stage B tile (16x64) interleaved by K-pairs ----
        {
            const float* brow0 = B + (size_t)(k0 + 2 * bpr)     * ldb + bn + bc;
            const float* brow1 = B + (size_t)(k0 + 2 * bpr + 1) * ldb + bn + bc;
            float4 g0 = *(const float4*)brow0;
            float4 g1 = *(const float4*)brow1;
            float* dst = Blds + bpr * B_PPITCH + bc * 2;
            ((float2*)dst)[0] = make_float2(g0.x, g1.x);
            ((float2*)dst)[1] = make_float2(g0.y, g1.y);
            ((float2*)dst)[2] = make_float2(g0.z, g1.z);
            ((float2*)dst)[3] = make_float2(g0.w, g1.w);
        }
        // Prefetch next tiles while this one is consumed (global_prefetch_b8)
        if (k0 + BK < K) {
            __builtin_prefetch(A + (size_t)(bm + ar0) * lda + k0 + BK + ac, 0, 1);
            __builtin_prefetch(B + (size_t)(k0 + BK + 2 * bpr) * ldb + bn + bc, 0, 1);
        }
        __syncthreads();

        #pragma unroll
        for (int kk = 0; kk < BK; kk += 4) {
            // A fragment: contiguous (k, k+1) pair -> one b64
            v2f a = *(const v2f*)(Alds + (wave * 16 + mf) * A_PITCH + kk + kp);
            // B fragments: interleaved pair at [pairrow][n*2] -> one b64 each
            const float* bbase = Blds + ((kk + kp) >> 1) * B_PPITCH + mf * 2;
            #pragma unroll
            for (int nt = 0; nt < 4; ++nt) {
                v2f b = *(const v2f*)(bbase + nt * 32);
                acc[nt] = __builtin_amdgcn_wmma_f32_16x16x4_f32(
                    false, a, false, b, (short)0, acc[nt], false, false);
            }
        }
        __syncthreads();
    }

    // Epilogue. C/D frag layout: vgpr v -> row v (+8 for lanes 16-31), lane&15 -> col.
    const int mrow0 = bm + wave * 16 + ((lane >> 4) << 3);
    #pragma unroll
    for (int nt = 0; nt < 4; ++nt) {
        int n = bn + nt * 16 + mf;
        #pragma unroll
        for (int v = 0; v < 8; ++v) {
            int m = mrow0 + v;
            float val = alpha * acc[nt][v];
            if (P)    val += beta * P[(size_t)m * N + n];
            if (bias) val += bias[n];
            if (relu) val = fmaxf(val, 0.f);
            C[(size_t)m * N + n] = val;
        }
    }
}

// x (64, 4096) -> z0 (4096, 64) : z0[u*64+n] = x[n*4096+u]
__global__ void k_transpose(const float* __restrict__ x, float* __restrict__ z)
{
    int idx = blockIdx.x * 256 + threadIdx.x;   // total 262144
    int n = idx & 63;
    int u = idx >> 6;
    z[idx] = x[(size_t)n * 4096 + u];
}

// Layer-1 linear (25 -> 32) + relu + maxpool(4).
// z: 25 buffers of (4096 x 64). h1[u'*2048 + n*32 + f]
__global__ __launch_bounds__(256)
void k_l1_pool(const float* __restrict__ z, const float* __restrict__ W1,
               const float* __restrict__ b1, float* __restrict__ h1)
{
    __shared__ float w[25 * 32];
    __shared__ float bb[32];
    for (int i = threadIdx.x; i < 800; i += 256) w[i] = W1[i];
    if (threadIdx.x < 32) bb[threadIdx.x] = b1[threadIdx.x];
    __syncthreads();

    int idx = blockIdx.x * 256 + threadIdx.x;   // total 1024*64*32 = 2097152
    int f  = idx & 31;
    int n  = (idx >> 5) & 63;
    int up = idx >> 11;
    float best = -3.4e38f;
    for (int j = 0; j < 4; ++j) {
        int u = up * 4 + j;
        const float* zp = z + (size_t)u * 64 + n;
        float s = bb[f];
        #pragma unroll
        for (int k = 0; k < 25; ++k) s += zp[(size_t)k * 262144] * w[k * 32 + f];
        best = fmaxf(best, s);
    }
    h1[idx] = fmaxf(best, 0.f);
}

// Layer-2 per-k linear accumulation: h2acc[u*4096+n*64+f] (+)= G_k @ W2[:,k-slice]
__global__ __launch_bounds__(256)
void k_l2_accum(const float* __restrict__ g, const float* __restrict__ W2,
                float* __restrict__ h2acc, int k, int init)
{
    __shared__ float w[32 * 64];    // w[fp*64+f] = W2[(fp*25+k)*64+f]
    for (int i = threadIdx.x; i < 2048; i += 256) {
        int fp = i >> 6, f = i & 63;
        w[i] = W2[(size_t)(fp * 25 + k) * 64 + f];
    }
    __syncthreads();

    int idx = blockIdx.x * 256 + threadIdx.x;   // total 1024*64*64 = 4194304
    int f = idx & 63;
    int n = (idx >> 6) & 63;
    int u = idx >> 12;
    const float* gp = g + (size_t)u * 2048 + n * 32;
    float s = 0.f;
    #pragma unroll
    for (int fp = 0; fp < 32; ++fp) s += gp[fp] * w[fp * 64 + f];
    if (init) h2acc[idx] = s;
    else      h2acc[idx] += s;
}

// bias + relu + maxpool(4) over nodes: h2[n*16384 + u'*64 + f]
__global__ void k_l2_pool(const float* __restrict__ h2acc, const float* __restrict__ b2,
                          float* __restrict__ h2)
{
    int idx = blockIdx.x * 256 + threadIdx.x;   // total 64*256*64 = 1048576
    int f  = idx & 63;
    int up = (idx >> 6) & 255;
    int n  = idx >> 14;
    float m = -3.4e38f;
    for (int j = 0; j < 4; ++j) {
        int u = up * 4 + j;
        m = fmaxf(m, h2acc[(size_t)u * 4096 + n * 64 + f]);
    }
    h2[idx] = fmaxf(m + b2[f], 0.f);
}

// Final tiny GEMM: (64 x 512) @ (512 x 10) + bo
__global__ void k_fc2(const float* __restrict__ f1, const float* __restrict__ Wo,
                      const float* __restrict__ bo, float* __restrict__ out)
{
    int idx = blockIdx.x * 64 + threadIdx.x;
    if (idx >= 640) return;
    int j = idx % 10, i = idx / 10;
    float s = bo[j];
    for (int c = 0; c < 512; ++c) s += f1[(size_t)i * 512 + c] * Wo[(size_t)c * 10 + j];
    out[idx] = s;
}

extern "C" void kernel_launch(void* const* d_in, const int* in_sizes, int n_in,
                              void* d_out, int out_size, void* d_ws, size_t ws_size,
                              hipStream_t stream)
{
    const float* x  = (const float*)d_in[0];
    const float* L0 = (const float*)d_in[1];
    const float* L1 = (const float*)d_in[2];
    const float* W1 = (const float*)d_in[3];
    const float* b1 = (const float*)d_in[4];
    const float* W2 = (const float*)d_in[5];
    const float* b2 = (const float*)d_in[6];
    const float* Wh = (const float*)d_in[7];
    const float* bh = (const float*)d_in[8];
    const float* Wo = (const float*)d_in[9];
    const float* bo = (const float*)d_in[10];

    // Workspace layout (floats). Total 13,926,400 floats = 55.7 MB.
    float* ws    = (float*)d_ws;
    float* z     = ws;                  // 25 x 262144 (phase 1 only)
    float* h1    = ws + 6553600;        // 1024 x 2048  (== G_0 for layer 2)
    float* g0    = ws;                  // reuse z region after phase 1
    float* g1    = ws + 2097152;
    float* h2acc = ws + 8650752;        // 1024 x 64 x 64
    float* h2    = ws + 12845056;       // 64 x 16384
    float* f1    = ws + 13893632;       // 64 x 512
    (void)ws_size; (void)in_sizes; (void)n_in; (void)out_size;

    // ---- Layer 1: Chebyshev on L0 (4096x4096), Z_k are (4096 x 64) ----
    k_transpose<<<1024, 256, 0, stream>>>(x, z);
    for (int k = 1; k < 25; ++k) {
        float*       zk = z + (size_t)k * 262144;
        const float* zc = z + (size_t)(k - 1) * 262144;
        const float* zp = (k >= 2) ? (z + (size_t)(k - 2) * 262144) : nullptr;
        float alpha = (k == 1) ? 1.f : 2.f;
        gemm_wmma_f32<<<dim3(1, 64), 128, 0, stream>>>(
            L0, 4096, zc, 64, zp, nullptr, zk, 4096, 64, 4096, alpha, -1.f, 0);
    }
    k_l1_pool<<<8192, 256, 0, stream>>>(z, W1, b1, h1);

    // ---- Layer 2: Chebyshev on L1 (1024x1024), G_k are (1024 x 2048) ----
    k_l2_accum<<<16384, 256, 0, stream>>>(h1, W2, h2acc, 0, 1);       // k=0, G_0 = h1
    gemm_wmma_f32<<<dim3(32, 16), 128, 0, stream>>>(                  // G_1 = L1 @ h1
        L1, 1024, h1, 2048, nullptr, nullptr, g0, 1024, 2048, 1024, 1.f, 0.f, 0);
    k_l2_accum<<<16384, 256, 0, stream>>>(g0, W2, h2acc, 1, 0);

    float* prevb = h1;   // G_{k-2}
    float* curb  = g0;   // G_{k-1}
    float* nxtb  = g1;   // G_k target
    for (int k = 2; k < 25; ++k) {
        // G_k = 2*L1@G_{k-1} - G_{k-2}; out may alias prev (elementwise safe)
        gemm_wmma_f32<<<dim3(32, 16), 128, 0, stream>>>(
            L1, 1024, curb, 2048, prevb, nullptr, nxtb, 1024, 2048, 1024, 2.f, -1.f, 0);
        k_l2_accum<<<16384, 256, 0, stream>>>(nxtb, W2, h2acc, k, 0);
        prevb = curb;
        curb  = nxtb;
        nxtb  = prevb;   // ping-pong: next output overwrites what becomes prev
    }
    k_l2_pool<<<4096, 256, 0, stream>>>(h2acc, b2, h2);

    // ---- FC head ----
    // f1 = relu(h2 (64x16384) @ Wh (16384x512) + bh)
    gemm_wmma_f32<<<dim3(8, 1), 128, 0, stream>>>(
        h2, 16384, Wh, 512, nullptr, bh, f1, 64, 512, 16384, 1.f, 0.f, 1);
    // out = f1 @ Wo + bo
    k_fc2<<<10, 64, 0, stream>>>(f1, Wo, bo, (float*)d_out);
}